// JoinAndSubsample_5695126634855
// MI455X (gfx1250) — compile-verified
//
#include <hip/hip_runtime.h>
#include <stdint.h>

// JoinAndSubsample: x(32,8192,80) f32 -> out(32,2731,560) f32
// out[b,t, ts*80+d] = x[b, clamp(3t + ts - 3, 0, 8191), d]
// Pure data movement (~280 MB -> ~12 us floor at 23.3 TB/s).
// CDNA5 path: async global->LDS->global DMA (ASYNCcnt-tracked), with
// non-temporal stores so the 196 MB write stream does not evict the
// 84 MB input from the 192 MB L2.

#define Bc       32
#define Tc       8192
#define FEATc    80
#define LEFTc    3
#define STRIDEc  3
#define NUM_TSc  7
#define TOUTc    2731            // (Tc-1)/STRIDEc + 1
#define OUTDc    560             // NUM_TSc*FEATc
#define TPBLK    32              // output rows per block
#define IN_ROWS  100             // (TPBLK-1)*STRIDEc + NUM_TSc
#define N_TILES  86              // ceil(TOUTc / TPBLK)
#define NLOAD    (IN_ROWS * (FEATc / 4))   // 2000 b128 copies
#define NSTORE   (TPBLK * (OUTDc / 4))     // 4480 b128 stores

__global__ __launch_bounds__(256)
void join_subsample_async(const float* __restrict__ x, float* __restrict__ out)
{
    __shared__ float smem[IN_ROWS * FEATc];   // 32 KB

    const int tid  = (int)threadIdx.x;
    const int tile = (int)blockIdx.x;         // 0..85
    const int b    = (int)blockIdx.y;         // 0..31
    const int t0   = tile * TPBLK;
    const int s0   = t0 * STRIDEc - LEFTc;    // first covering input row (may be <0)

    // Low 32 bits of a generic LDS pointer == LDS byte offset (aperture rule).
    const uint32_t lds_base = (uint32_t)(uintptr_t)&smem[0];

    // ---- Phase 1: DMA global -> LDS (edge-clamped at the source address) ----
    const float* xb = x + (size_t)b * Tc * FEATc;
    for (int c = tid; c < NLOAD; c += 256) {
        int row = c / (FEATc / 4);
        int v4  = c - row * (FEATc / 4);
        int gr  = s0 + row;
        gr = gr < 0 ? 0 : (gr > (Tc - 1) ? (Tc - 1) : gr);
        uint64_t src  = (uint64_t)(uintptr_t)(xb + (size_t)gr * FEATc + (size_t)v4 * 4);
        uint32_t ldst = lds_base + (uint32_t)(row * FEATc + v4 * 4) * 4u;
        asm volatile("global_load_async_to_lds_b128 %0, %1, off"
                     :: "v"(ldst), "v"(src) : "memory");
    }
    asm volatile("s_wait_asynccnt 0x0" ::: "memory");
    __syncthreads();

    // ---- Phase 2: DMA LDS -> global, non-temporal (write-once stream) ----
    float* ob = out + (size_t)b * TOUTc * OUTDc;
    for (int c = tid; c < NSTORE; c += 256) {
        int tr = c / (OUTDc / 4);             // local output row 0..31
        int j4 = c - tr * (OUTDc / 4);        // vec4 column 0..139
        int t  = t0 + tr;
        if (t >= TOUTc) continue;
        int ts = j4 / (FEATc / 4);            // which of the 7 window rows
        int d4 = j4 - ts * (FEATc / 4);
        uint32_t ldsrc = lds_base + (uint32_t)((tr * STRIDEc + ts) * FEATc + d4 * 4) * 4u;
        uint64_t dst   = (uint64_t)(uintptr_t)(ob + (size_t)t * OUTDc + (size_t)j4 * 4);
        asm volatile("global_store_async_from_lds_b128 %0, %1, off th:TH_STORE_NT"
                     :: "v"(dst), "v"(ldsrc) : "memory");
    }
    // s_endpgm performs an implicit wait-idle, draining outstanding async stores.
}

extern "C" void kernel_launch(void* const* d_in, const int* in_sizes, int n_in,
                              void* d_out, int out_size, void* d_ws, size_t ws_size,
                              hipStream_t stream)
{
    (void)in_sizes; (void)n_in; (void)d_ws; (void)ws_size; (void)out_size;
    const float* x = (const float*)d_in[0];
    float* out = (float*)d_out;
    dim3 grid(N_TILES, Bc, 1);
    dim3 block(256, 1, 1);
    join_subsample_async<<<grid, block, 0, stream>>>(x, out);
}